// FPSCoarsening_20899310862523
// MI455X (gfx1250) — compile-verified
//
#include <hip/hip_runtime.h>
#include <math.h>

// ---------------- problem constants ----------------
#define N_PTS   32768
#define M_SEEDS 512
#define D_FEAT  128
#define C_CLS   20
#define K_NB    32
#define EPSV    1e-6f
#define R2_BALL (0.15f * 0.15f)

// GEMM config: X = [p(3) | s(20) | h(128) | ones(1) @ col 151 | zero pad] -> 160 cols
#define GCOLS   160
#define KSPLIT  16
#define KCHUNK  (N_PTS / KSPLIT)   // 2048

// ---------------- output layout (float elements) ----------------
#define O_PSUPER 0          // 512*3
#define O_RSUPER 1536       // 512*3
#define O_SSUPER 3072       // 512*20
#define O_EDGE   13312      // 2*512*32
#define O_S      46080      // 32768*512
#define O_HA     16823296   // 32768
#define O_HP     16856064   // 512*128

// ---------------- workspace layout (bytes) ----------------
#define W_SEEDS  0          // 512 int
#define W_PSEED  2048       // 512*3 f
#define W_RMEAN  8192       // 512 f
#define W_SEGMIN 10240      // 512*3 int
#define W_SEGMAX 16384      // 512*3 int
#define W_CNT    22528      // 512 int
#define W_CONST  24576      // 2 f : {1.0f, 0.0f} for stride-0 B columns
#define W_G      24832      // KSPLIT*512*160 f  (~5.24 MB)

typedef __attribute__((ext_vector_type(2))) float v2f;
typedef __attribute__((ext_vector_type(8))) float v8f;

// ============================================================
// 1) Farthest point sampling: 1 workgroup, 1024 threads.
//    Point coords in registers (32 pts/lane), mind[] in LDS (128KB).
// ============================================================
extern "C" __global__ __launch_bounds__(1024)
void fps_kernel(const float* __restrict__ p, int* __restrict__ seeds) {
  extern __shared__ float lds[];
  float* mind  = lds;                       // N_PTS floats
  float* rv    = lds + N_PTS;               // 1024 floats
  int*   ri    = (int*)(rv + 1024);         // 1024 ints
  int*   lastp = ri + 1024;                 // 1 int

  const int tid = threadIdx.x;
  float px[32], py[32], pz[32];
#pragma unroll
  for (int t = 0; t < 32; ++t) {
    int i = tid + t * 1024;
    px[t] = p[3*i+0]; py[t] = p[3*i+1]; pz[t] = p[3*i+2];
    mind[i] = 3.4e38f;
  }
  if (tid == 0) { seeds[0] = 0; *lastp = 0; }
  __syncthreads();

  for (int step = 1; step < M_SEEDS; ++step) {
    int last = *lastp;
    float lx = p[3*last+0], ly = p[3*last+1], lz = p[3*last+2];
    float best = -1.0f; int bidx = 0;
#pragma unroll
    for (int t = 0; t < 32; ++t) {
      int i = tid + t * 1024;
      float dx = px[t]-lx, dy = py[t]-ly, dz = pz[t]-lz;
      float d = dx*dx + dy*dy + dz*dz;
      float m = fminf(mind[i], d);
      mind[i] = m;
      if (m > best) { best = m; bidx = i; }   // ascending i -> first max kept
    }
    rv[tid] = best; ri[tid] = bidx;
    __syncthreads();
    for (int off = 512; off > 0; off >>= 1) {
      if (tid < off) {
        float ov = rv[tid+off]; int oi = ri[tid+off];
        float mv = rv[tid];     int mi = ri[tid];
        if (ov > mv || (ov == mv && oi < mi)) { rv[tid] = ov; ri[tid] = oi; }
      }
      __syncthreads();
    }
    if (tid == 0) { seeds[step] = ri[0]; *lastp = ri[0]; }
    __syncthreads();
  }
}

// ============================================================
// 2) Gather seed coords / radii mean; init segment min/max/counts
//    and the stride-0 constant block for the GEMM B matrix.
// ============================================================
extern "C" __global__ void gather_kernel(const float* __restrict__ p,
                                         const float* __restrict__ r,
                                         const int* __restrict__ seeds,
                                         float* __restrict__ p_seeds,
                                         float* __restrict__ r_mean,
                                         int* __restrict__ seg_min,
                                         int* __restrict__ seg_max,
                                         int* __restrict__ cnt,
                                         float* __restrict__ cst) {
  int m = blockIdx.x * blockDim.x + threadIdx.x;
  if (m >= M_SEEDS) return;
  if (m == 0) { cst[0] = 1.0f; cst[1] = 0.0f; }
  int sidx = seeds[m];
  p_seeds[3*m+0] = p[3*sidx+0];
  p_seeds[3*m+1] = p[3*sidx+1];
  p_seeds[3*m+2] = p[3*sidx+2];
  r_mean[m] = (r[3*sidx+0] + r[3*sidx+1] + r[3*sidx+2]) * (1.0f/3.0f) + EPSV;
#pragma unroll
  for (int d = 0; d < 3; ++d) {
    seg_min[3*m+d] = 0x7f800000;  // +inf bits (p >= 0 so int-order == float-order)
    seg_max[3*m+d] = 0;           // 0.0f bits
  }
  cnt[m] = 0;
}

// ============================================================
// 3) Per-point: dist to 512 seeds, MLP scores, row softmax -> S,
//    argmin -> hard_assign, bbox via deterministic int atomics.
//    One block (128 thr) per point; 4 columns per thread.
// ============================================================
extern "C" __global__ __launch_bounds__(128)
void assign_kernel(const float* __restrict__ p,
                   const float* __restrict__ w1, const float* __restrict__ b1,
                   const float* __restrict__ w2, const float* __restrict__ b2,
                   const float* __restrict__ p_seeds,
                   const float* __restrict__ r_mean,
                   float* __restrict__ out,
                   int* __restrict__ seg_min, int* __restrict__ seg_max,
                   int* __restrict__ cnt) {
  __shared__ float sw1[16], sb1[16], sw2[16];
  __shared__ float sb2;
  __shared__ float redf[128];
  __shared__ int   redi[128];

  const int i = blockIdx.x;
  const int tid = threadIdx.x;
  if (tid < 16) { sw1[tid] = w1[tid]; sb1[tid] = b1[tid]; sw2[tid] = w2[tid]; }
  if (tid == 0) sb2 = b2[0];
  __syncthreads();

  const float pxi = p[3*i+0], pyi = p[3*i+1], pzi = p[3*i+2];
  float score[4];
  float mind = 3.4e38f; int minc = 0;
#pragma unroll
  for (int t = 0; t < 4; ++t) {
    int c = t * 128 + tid;
    float dx = pxi - p_seeds[3*c+0];
    float dy = pyi - p_seeds[3*c+1];
    float dz = pzi - p_seeds[3*c+2];
    float d2 = dx*dx + dy*dy + dz*dz;
    float dist = sqrtf(fmaxf(d2, 0.0f));
    if (dist < mind) { mind = dist; minc = c; }  // ascending c -> first min kept
    float dn = dist / r_mean[c];
    float sc = sb2;
#pragma unroll
    for (int k = 0; k < 16; ++k) {
      float hv = fmaxf(fmaf(dn, sw1[k], sb1[k]), 0.0f);
      sc = fmaf(hv, sw2[k], sc);
    }
    score[t] = sc;
  }

  // argmin reduction (value, index) for hard_assign
  redf[tid] = mind; redi[tid] = minc;
  __syncthreads();
  for (int off = 64; off > 0; off >>= 1) {
    if (tid < off) {
      float ov = redf[tid+off]; int oi = redi[tid+off];
      if (ov < redf[tid] || (ov == redf[tid] && oi < redi[tid])) {
        redf[tid] = ov; redi[tid] = oi;
      }
    }
    __syncthreads();
  }
  int assign = redi[0];
  if (tid == 0) {
    out[O_HA + i] = (float)assign;
    atomicMin(&seg_min[3*assign+0], __float_as_int(pxi));
    atomicMin(&seg_min[3*assign+1], __float_as_int(pyi));
    atomicMin(&seg_min[3*assign+2], __float_as_int(pzi));
    atomicMax(&seg_max[3*assign+0], __float_as_int(pxi));
    atomicMax(&seg_max[3*assign+1], __float_as_int(pyi));
    atomicMax(&seg_max[3*assign+2], __float_as_int(pzi));
    atomicAdd(&cnt[assign], 1);
  }

  // softmax: row max
  float lmax = fmaxf(fmaxf(score[0], score[1]), fmaxf(score[2], score[3]));
  __syncthreads();
  redf[tid] = lmax;
  __syncthreads();
  for (int off = 64; off > 0; off >>= 1) {
    if (tid < off) redf[tid] = fmaxf(redf[tid], redf[tid+off]);
    __syncthreads();
  }
  float gmax = redf[0];
  __syncthreads();
  // row sum of exp
  float lsum = 0.0f;
#pragma unroll
  for (int t = 0; t < 4; ++t) lsum += expf(score[t] - gmax);
  redf[tid] = lsum;
  __syncthreads();
  for (int off = 64; off > 0; off >>= 1) {
    if (tid < off) redf[tid] += redf[tid+off];
    __syncthreads();
  }
  float inv = 1.0f / redf[0];
#pragma unroll
  for (int t = 0; t < 4; ++t) {
    int c = t * 128 + tid;
    out[(size_t)O_S + (size_t)i * M_SEEDS + c] = expf(score[t] - gmax) * inv;
  }
}

// ============================================================
// 4) Pooling GEMM partials via fp32 WMMA (V_WMMA_F32_16X16X4_F32).
//    Each wave: one 16-row m-tile x TWO 16-col n-tiles (A reuse),
//    one K-split of 2048. Branch-free B via stride-0 const columns;
//    pure pointer-increment addressing; 1-deep software pipeline.
// ============================================================
#define WMMA_F32(A, B, C) \
  __builtin_amdgcn_wmma_f32_16x16x4_f32(false, (A), false, (B), (short)0, (C), false, false)

extern "C" __global__ __launch_bounds__(32)
void pool_wmma_kernel(const float* __restrict__ p,
                      const float* __restrict__ s,
                      const float* __restrict__ h,
                      const float* __restrict__ cst,
                      const float* __restrict__ S,
                      float* __restrict__ G) {
  const int b   = blockIdx.x;
  const int mt  = b % 32;           // 32 m-tiles (512/16)
  const int ng  = (b / 32) % 5;     // 5 n-groups of 2 tiles (160/32)
  const int ks  = b / 160;          // 16 k-splits
  const int lane  = threadIdx.x;
  const int col16 = lane & 15;
  const int krow  = (lane >> 4) << 1;   // lanes 0-15 -> K+0/K+1, lanes 16-31 -> K+2/K+3
  const int m0 = mt * 16;
  const int kbeg = ks * KCHUNK;

  const int n0 = (ng * 2 + 0) * 16 + col16;
  const int n1 = (ng * 2 + 1) * 16 + col16;

  // per-lane B column sources; constant columns use stride-0 pointers (no branch in loop)
  const float* b0base; int st0;
  if      (n0 < 3)   { b0base = p + n0;        st0 = 3; }
  else if (n0 < 23)  { b0base = s + (n0 - 3);  st0 = 20; }
  else if (n0 < 151) { b0base = h + (n0 - 23); st0 = 128; }
  else if (n0 == 151){ b0base = cst;           st0 = 0; }    // ones -> column sums
  else               { b0base = cst + 1;       st0 = 0; }    // zero pad
  const float* b1base; int st1;
  if      (n1 < 3)   { b1base = p + n1;        st1 = 3; }
  else if (n1 < 23)  { b1base = s + (n1 - 3);  st1 = 20; }
  else if (n1 < 151) { b1base = h + (n1 - 23); st1 = 128; }
  else if (n1 == 151){ b1base = cst;           st1 = 0; }
  else               { b1base = cst + 1;       st1 = 0; }

  const float* ap  = S + (size_t)(kbeg + krow) * M_SEEDS + (m0 + col16);
  const float* bp0 = b0base + (size_t)(kbeg + krow) * st0;
  const float* bq0 = bp0 + st0;                 // row kr+1
  const float* bp1 = b1base + (size_t)(kbeg + krow) * st1;
  const float* bq1 = bp1 + st1;
  const int astep = 4 * M_SEEDS;
  const int bstep0 = 4 * st0, bstep1 = 4 * st1;

  v8f acc0 = {}, acc1 = {};
  v2f a, b0, b1;
  a.x  = ap[0];  a.y  = ap[M_SEEDS];            // imm-offset 2048B for K+1 row
  b0.x = bp0[0]; b0.y = bq0[0];
  b1.x = bp1[0]; b1.y = bq1[0];

  const int iters = KCHUNK / 4;                 // 512
  for (int it = 0; it < iters - 1; ++it) {
    ap  += astep;
    bp0 += bstep0; bq0 += bstep0;
    bp1 += bstep1; bq1 += bstep1;
    v2f an, b0n, b1n;
    an.x  = ap[0];  an.y  = ap[M_SEEDS];
    b0n.x = bp0[0]; b0n.y = bq0[0];
    b1n.x = bp1[0]; b1n.y = bq1[0];
    acc0 = WMMA_F32(a, b0, acc0);
    acc1 = WMMA_F32(a, b1, acc1);
    a = an; b0 = b0n; b1 = b1n;
  }
  acc0 = WMMA_F32(a, b0, acc0);
  acc1 = WMMA_F32(a, b1, acc1);

  // store 16x16 partials: VGPR v -> row m0+v (lanes<16) / m0+8+v (lanes>=16)
  float* Gp = G + (size_t)ks * M_SEEDS * GCOLS;
  const int rbase = m0 + ((lane >> 4) << 3);
#pragma unroll
  for (int v = 0; v < 8; ++v) {
    Gp[(size_t)(rbase + v) * GCOLS + n0] = acc0[v];
    Gp[(size_t)(rbase + v) * GCOLS + n1] = acc1[v];
  }
}

// ============================================================
// 5) Finalize: reduce K-splits, divide by (colsum + eps),
//    scatter into p_super/s_super/h_pooled; compute r_super.
// ============================================================
extern "C" __global__ __launch_bounds__(192)
void finalize_kernel(const float* __restrict__ G,
                     const int* __restrict__ seg_min,
                     const int* __restrict__ seg_max,
                     const int* __restrict__ cnt,
                     float* __restrict__ out) {
  __shared__ float sh[GCOLS];
  const int m = blockIdx.x;
  const int n = threadIdx.x;
  if (n < GCOLS) {
    float acc = 0.0f;
    for (int ks = 0; ks < KSPLIT; ++ks)
      acc += G[((size_t)ks * M_SEEDS + m) * GCOLS + n];
    sh[n] = acc;
  }
  __syncthreads();
  const float denom = sh[151] + EPSV;
  if (n < 3) {
    out[O_PSUPER + 3*m + n] = sh[n] / denom;
    int c = cnt[m];
    float mn = __int_as_float(seg_min[3*m+n]);
    float mx = __int_as_float(seg_max[3*m+n]);
    out[O_RSUPER + 3*m + n] = (c > 0) ? (mx - mn) * 0.5f : 0.0f;
  } else if (n < 23) {
    out[O_SSUPER + 20*m + (n - 3)] = sh[n] / denom;
  } else if (n < 151) {
    out[(size_t)O_HP + 128*m + (n - 23)] = sh[n] / denom;
  }
}

// ============================================================
// 6) Radius graph on p_super (512 pts): per-center stable 32-NN
//    within radius; -1 fill. Matches lax.top_k tie semantics.
// ============================================================
extern "C" __global__ __launch_bounds__(128)
void radius_kernel(const float* __restrict__ psuper, float* __restrict__ out) {
  __shared__ float spx[M_SEEDS], spy[M_SEEDS], spz[M_SEEDS];
  __shared__ float d2s[M_SEEDS];
  __shared__ float redf[128];
  __shared__ int   redi[128];
  __shared__ float selv[K_NB];
  __shared__ int   seli[K_NB];

  const int m = blockIdx.x, tid = threadIdx.x;
  for (int t = tid; t < M_SEEDS; t += 128) {
    spx[t] = psuper[3*t+0]; spy[t] = psuper[3*t+1]; spz[t] = psuper[3*t+2];
  }
  __syncthreads();
  const float cx = spx[m], cy = spy[m], cz = spz[m];
  for (int t = tid; t < M_SEEDS; t += 128) {
    float dx = spx[t]-cx, dy = spy[t]-cy, dz = spz[t]-cz;
    float d2 = dx*dx + dy*dy + dz*dz;
    d2s[t] = (t == m) ? 3.4e38f : d2;   // no self-loop
  }
  __syncthreads();

  for (int slot = 0; slot < K_NB; ++slot) {
    float bv = 3.5e38f; int bi = 0;
    for (int t = tid; t < M_SEEDS; t += 128) {
      float v = d2s[t];
      if (v < bv) { bv = v; bi = t; }
    }
    redf[tid] = bv; redi[tid] = bi;
    __syncthreads();
    for (int off = 64; off > 0; off >>= 1) {
      if (tid < off) {
        float ov = redf[tid+off]; int oi = redi[tid+off];
        if (ov < redf[tid] || (ov == redf[tid] && oi < redi[tid])) {
          redf[tid] = ov; redi[tid] = oi;
        }
      }
      __syncthreads();
    }
    if (tid == 0) { selv[slot] = redf[0]; seli[slot] = redi[0]; d2s[redi[0]] = 3.4e38f; }
    __syncthreads();
  }

  if (tid < K_NB) {
    int valid = selv[tid] <= R2_BALL;
    int src = valid ? seli[tid] : -1;
    int dst = valid ? m : -1;
    out[O_EDGE + m * K_NB + tid]                  = (float)src;
    out[O_EDGE + M_SEEDS * K_NB + m * K_NB + tid] = (float)dst;
  }
}

// ============================================================
extern "C" void kernel_launch(void* const* d_in, const int* in_sizes, int n_in,
                              void* d_out, int out_size, void* d_ws, size_t ws_size,
                              hipStream_t stream) {
  const float* p  = (const float*)d_in[0];
  const float* r  = (const float*)d_in[1];
  const float* s  = (const float*)d_in[2];
  const float* h  = (const float*)d_in[3];
  const float* w1 = (const float*)d_in[4];
  const float* b1 = (const float*)d_in[5];
  const float* w2 = (const float*)d_in[6];
  const float* b2 = (const float*)d_in[7];
  float* out = (float*)d_out;
  char*  ws  = (char*)d_ws;

  int*   seeds   = (int*)(ws + W_SEEDS);
  float* p_seeds = (float*)(ws + W_PSEED);
  float* r_mean  = (float*)(ws + W_RMEAN);
  int*   seg_min = (int*)(ws + W_SEGMIN);
  int*   seg_max = (int*)(ws + W_SEGMAX);
  int*   cnt     = (int*)(ws + W_CNT);
  float* cst     = (float*)(ws + W_CONST);
  float* G       = (float*)(ws + W_G);

  // 1) FPS (one WGP; 128KB mind[] + reduction scratch in dynamic LDS)
  size_t fps_lds = (size_t)N_PTS * 4 + 1024 * 4 + 1024 * 4 + 16;
  hipLaunchKernelGGL(fps_kernel, dim3(1), dim3(1024), fps_lds, stream, p, seeds);

  // 2) seed gather + segment/const init
  hipLaunchKernelGGL(gather_kernel, dim3(2), dim3(256), 0, stream,
                     p, r, seeds, p_seeds, r_mean, seg_min, seg_max, cnt, cst);

  // 3) S / hard_assign / bbox atomics
  hipLaunchKernelGGL(assign_kernel, dim3(N_PTS), dim3(128), 0, stream,
                     p, w1, b1, w2, b2, p_seeds, r_mean, out, seg_min, seg_max, cnt);

  // 4) fp32 WMMA pooling GEMM partials (S.T @ [p|s|h|1])
  hipLaunchKernelGGL(pool_wmma_kernel, dim3(32 * 5 * KSPLIT), dim3(32), 0, stream,
                     p, s, h, cst, out + O_S, G);

  // 5) reduce + normalize + r_super
  hipLaunchKernelGGL(finalize_kernel, dim3(M_SEEDS), dim3(192), 0, stream,
                     G, seg_min, seg_max, cnt, out);

  // 6) radius graph on p_super
  hipLaunchKernelGGL(radius_kernel, dim3(M_SEEDS), dim3(128), 0, stream,
                     out + O_PSUPER, out);
}